// _TemplateProposalLayer_73718818669043
// MI455X (gfx1250) — compile-verified
//
#include <hip/hip_runtime.h>
#include <stdint.h>

#define B_      16
#define N_      65536
#define G_      64
#define M_      (N_ + G_)
#define R_      128
#define NBUCK   1024
#define PSTRIDE 1028          /* prefix stride (ints): 4112 B, 16B aligned  */
#define FG_TH   0.7f
#define BG_HI   0.3f

typedef unsigned int u32x4 __attribute__((ext_vector_type(4)));
typedef unsigned int u32x8 __attribute__((ext_vector_type(8)));

// ---------------------------------------------------------------------------
// CDNA5 Tensor Data Mover: 1-line tile (nelem x 1) of 4-byte data, global->LDS.
// D# layout per CDNA5 ISA; 2-group form (<=2D).
// Issue from one wave only; completion via TENSORcnt.
// ---------------------------------------------------------------------------
__device__ __forceinline__ void tdm_load_words_to_lds(void* lds_ptr,
                                                      const void* gptr,
                                                      unsigned nelem /* dwords */) {
    unsigned lds_off = (unsigned)(uintptr_t)lds_ptr;       // LDS byte offset
    unsigned long long ga = (unsigned long long)(uintptr_t)gptr;

    u32x4 g0;
    g0[0] = 0x1u;                                          // count=1 (valid), user D#
    g0[1] = lds_off;                                       // lds_addr [63:32]
    g0[2] = (unsigned)(ga & 0xffffffffu);                  // global_addr[31:0]
    g0[3] = (unsigned)((ga >> 32) & 0x01ffffffu)           // global_addr[56:32]
          | (2u << 30);                                    // type=2 ("image")

    u32x8 g1;
    g1[0] = 0x2u << 16;                                    // wg_mask=0, data_size=2 (4B)
    g1[1] = (nelem & 0xffffu) << 16;                       // tensor_dim0[15:0] @bits[63:48]
    g1[2] = ((nelem >> 16) & 0xffffu) | (1u << 16);        // tensor_dim0[31:16], tensor_dim1=1
    g1[3] = (nelem & 0xffffu) << 16;                       // tensor_dim1 hi=0, tile_dim0=nelem
    g1[4] = 1u;                                            // tile_dim1=1, tile_dim2=0
    g1[5] = nelem;                                         // tensor_dim0_stride[31:0]
    g1[6] = (nelem & 0xffffu) << 16;                       // stride0 hi=0, stride1[15:0]
    g1[7] = 0u;                                            // stride1 hi

    asm volatile("tensor_load_to_lds %0, %1" :: "s"(g0), "s"(g1) : "memory");
}

// ---------------------------------------------------------------------------
// CDNA5 async global->LDS copy (16B per lane), tracked by ASYNCcnt.
// ---------------------------------------------------------------------------
__device__ __forceinline__ void async_copy16_to_lds(void* lds_ptr, const void* gptr) {
    unsigned lds_off = (unsigned)(uintptr_t)lds_ptr;
    unsigned long long ga = (unsigned long long)(uintptr_t)gptr;
    asm volatile("global_load_async_to_lds_b128 %0, %1, off"
                 :: "v"(lds_off), "v"(ga)
                 : "memory");
}
__device__ __forceinline__ void wait_async0() {
    asm volatile("s_wait_asynccnt 0" ::: "memory");
}

// ---------------------------------------------------------------------------
// Kernel 0: zero the histogram + bucket-fill cursors (per-launch, replay-safe).
// ---------------------------------------------------------------------------
__global__ void k_zero(int* __restrict__ hist, int* __restrict__ fill, int n) {
    int i = blockIdx.x * blockDim.x + threadIdx.x;
    if (i < n) { hist[i] = 0; fill[i] = 0; }
}

// ---------------------------------------------------------------------------
// Kernel 1: IoU max/argmax per ROI + FG/BG classification + key histogram.
// GT rows staged into LDS via the Tensor Data Mover.
// ---------------------------------------------------------------------------
__global__ void k_overlap(const float* __restrict__ all_rois,
                          const float* __restrict__ gt_boxes,
                          const float* __restrict__ u_perm,
                          float* __restrict__ max_ov,
                          int*   __restrict__ gt_assign,
                          int*   __restrict__ hist) {
    __shared__ __align__(16) float sGT[G_ * 6];   // raw gt rows (x1,y1,x2,y2,label,tid)
    __shared__ float sArea[G_];
    __shared__ int   sZero[G_];
    __shared__ int   sHist[2 * NBUCK];

    const int b = blockIdx.y;
    const int t = threadIdx.x;

    for (int k = t; k < 2 * NBUCK; k += blockDim.x) sHist[k] = 0;

    // TDM: DMA 64*6 = 384 dwords of GT data into LDS (wave 0 issues; EXEC ignored).
    if (t < warpSize) {
        tdm_load_words_to_lds(&sGT[0], gt_boxes + (size_t)b * G_ * 6, G_ * 6);
    }
    __builtin_amdgcn_s_wait_tensorcnt(0);
    __syncthreads();

    if (t < G_) {
        float gx1 = sGT[t * 6 + 0], gy1 = sGT[t * 6 + 1];
        float gx2 = sGT[t * 6 + 2], gy2 = sGT[t * 6 + 3];
        float gw = gx2 - gx1 + 1.0f, gh = gy2 - gy1 + 1.0f;
        sArea[t] = gw * gh;
        sZero[t] = (gw == 1.0f) && (gh == 1.0f);
    }
    __syncthreads();

    const int i = blockIdx.x * blockDim.x + t;
    if (i < M_) {
        float ax1, ay1, ax2, ay2;
        if (i < N_) {
            const float* p = all_rois + ((size_t)b * N_ + i) * 5;
            if (i + 8192 < N_)  // gfx1250 global_prefetch_b8
                __builtin_prefetch(all_rois + ((size_t)b * N_ + i + 8192) * 5, 0, 0);
            ax1 = p[1]; ay1 = p[2]; ax2 = p[3]; ay2 = p[4];
        } else {
            const float* p = &sGT[(i - N_) * 6];   // gt_append row = [0, gtb]
            ax1 = p[0]; ay1 = p[1]; ax2 = p[2]; ay2 = p[3];
        }
        float aw = ax2 - ax1 + 1.0f, ah = ay2 - ay1 + 1.0f;
        float areaA = aw * ah;
        bool  aZero = (aw == 1.0f) && (ah == 1.0f);

        float best = -3.0e38f;
        int   barg = 0;
        for (int g = 0; g < G_; ++g) {
            float gx1 = sGT[g * 6 + 0], gy1 = sGT[g * 6 + 1];
            float gx2 = sGT[g * 6 + 2], gy2 = sGT[g * 6 + 3];
            float iw = fminf(ax2, gx2) - fmaxf(ax1, gx1) + 1.0f; iw = fmaxf(iw, 0.0f);
            float ih = fminf(ay2, gy2) - fmaxf(ay1, gy1) + 1.0f; ih = fmaxf(ih, 0.0f);
            float inter = iw * ih;
            float ov = inter / (areaA + sArea[g] - inter);
            if (sZero[g]) ov = 0.0f;
            if (ov > best) { best = ov; barg = g; }   // first-argmax semantics
        }
        if (aZero) { best = -1.0f; barg = 0; }

        size_t idx = (size_t)b * M_ + i;
        max_ov[idx]    = best;
        gt_assign[idx] = barg;

        // fg: >=0.7 ; bg: <0.3 (ref mask ((ov<0.3)&(ov>=0))|(ov<0) == ov<0.3)
        int set = -1;
        if (best >= FG_TH)       set = 0;
        else if (best < BG_HI)   set = 1;
        if (set >= 0) {
            float key = u_perm[idx];
            int bk = (int)(key * (float)NBUCK);
            bk = bk < 0 ? 0 : (bk > NBUCK - 1 ? NBUCK - 1 : bk);
            atomicAdd(&sHist[set * NBUCK + bk], 1);
        }
    }
    __syncthreads();
    for (int k = t; k < 2 * NBUCK; k += blockDim.x) {
        int v = sHist[k];
        if (v) atomicAdd(&hist[((size_t)b * 2) * NBUCK + k], v);
    }
}

// ---------------------------------------------------------------------------
// Kernel 2: exclusive prefix over 1024 bins, per (batch, set).
// ---------------------------------------------------------------------------
__global__ void k_prefix(const int* __restrict__ hist, int* __restrict__ prefix) {
    int bs = blockIdx.x;                 // 0..B*2-1
    if (threadIdx.x != 0) return;
    const int* h = hist + (size_t)bs * NBUCK;
    int* p = prefix + (size_t)bs * PSTRIDE;
    int acc = 0;
    for (int k = 0; k < NBUCK; ++k) { p[k] = acc; acc += h[k]; }
    p[NBUCK] = acc;
    p[NBUCK + 1] = 0; p[NBUCK + 2] = 0; p[NBUCK + 3] = 0;   // pad (keeps LDS copy defined)
}

// ---------------------------------------------------------------------------
// Kernel 3: scatter (key, index) pairs into bucket-sorted storage.
// fg region [0, fgN), bg region [fgN, fgN+bgN) per batch.
// ---------------------------------------------------------------------------
__global__ void k_scatter(const float* __restrict__ u_perm,
                          const float* __restrict__ max_ov,
                          const int*   __restrict__ prefix,
                          int*   __restrict__ fill,
                          float* __restrict__ skey,
                          int*   __restrict__ sidx) {
    int b = blockIdx.y;
    int i = blockIdx.x * blockDim.x + threadIdx.x;
    if (i >= M_) return;
    size_t idx = (size_t)b * M_ + i;
    float mo = max_ov[idx];
    int set;
    if (mo >= FG_TH)      set = 0;
    else if (mo < BG_HI)  set = 1;
    else return;

    float key = u_perm[idx];
    int bk = (int)(key * (float)NBUCK);
    bk = bk < 0 ? 0 : (bk > NBUCK - 1 ? NBUCK - 1 : bk);

    const int* p  = prefix + ((size_t)b * 2 + set) * PSTRIDE;
    int fgTotal   = prefix[((size_t)b * 2) * PSTRIDE + NBUCK];
    int base      = (set == 1) ? fgTotal : 0;
    int pos       = atomicAdd(&fill[((size_t)b * 2 + set) * NBUCK + bk], 1);
    size_t dst    = (size_t)b * M_ + base + p[bk] + pos;
    skey[dst] = key;
    sidx[dst] = i;
}

// ---------------------------------------------------------------------------
// Rank-r selection: binary-search the (LDS-resident) prefix, then exact stable
// rank inside the bucket via (key, idx) comparisons (atomic-order agnostic).
// ---------------------------------------------------------------------------
__device__ int select_rank(const int* P,                    // LDS prefix (1025 entries)
                           const float* __restrict__ K,
                           const int* __restrict__ I,
                           size_t base, int r) {
    int lo = 0, hi = NBUCK;
    while (lo < hi) {                 // smallest k with P[k+1] > r
        int mid = (lo + hi) >> 1;
        if (P[mid + 1] <= r) lo = mid + 1; else hi = mid;
    }
    int s = P[lo], e = P[lo + 1];
    int target = r - s;
    for (int a = s; a < e; ++a) {
        float ka = K[base + a]; int ia = I[base + a];
        int cnt = 0;
        for (int c = s; c < e; ++c) {
            float kc = K[base + c]; int ic = I[base + c];
            cnt += (kc < ka) || (kc == ka && ic < ia);
        }
        if (cnt == target) return ia;
    }
    return 0;
}

// ---------------------------------------------------------------------------
// Kernel 4: per-batch sampling (all reference edge cases) + output gather.
// Prefix tables (8224 B) staged into LDS via async global->LDS DMA.
// ---------------------------------------------------------------------------
__global__ void k_sample(const float* __restrict__ all_rois,
                         const float* __restrict__ gt_boxes,
                         const float* __restrict__ u_fg,
                         const float* __restrict__ u_bg,
                         const int*   __restrict__ prefix,
                         const float* __restrict__ skey,
                         const int*   __restrict__ sidx,
                         const int*   __restrict__ gt_assign,
                         float* __restrict__ out) {
    __shared__ __align__(16) int sPfx[2 * PSTRIDE];   // [0..]: fg, [PSTRIDE..]: bg

    int b = blockIdx.x;
    int pos = threadIdx.x;

    // Async-DMA both prefix tables: 2*PSTRIDE*4 = 8224 B = 514 x b128 chunks.
    {
        const char* src = (const char*)(prefix + (size_t)b * 2 * PSTRIDE);
        char* dst = (char*)&sPfx[0];
        const int nchunk = (2 * PSTRIDE * 4) / 16;        // 514
        for (int c = pos; c < nchunk; c += blockDim.x)
            async_copy16_to_lds(dst + (size_t)c * 16, src + (size_t)c * 16);
    }
    wait_async0();
    __syncthreads();

    const int* pf = &sPfx[0];
    const int* pb = &sPfx[PSTRIDE];
    int fgN = pf[NBUCK], bgN = pb[NBUCK];

    bool both    = (fgN > 0) && (bgN > 0);
    bool only_fg = (fgN > 0) && (bgN == 0);
    int fg_this  = both ? (fgN < 32 ? fgN : 32) : (only_fg ? R_ : 0);
    bool is_fg   = pos < fg_this;

    int keep = 0;   // fgN==0 && bgN==0 -> stable argsort of all-equal keys -> index 0
    if (is_fg) {
        int r;
        if (only_fg) {
            r = (int)(u_fg[b * R_ + pos] * (float)fgN);   // trunc toward zero
            r = r < 0 ? 0 : (r > fgN - 1 ? fgN - 1 : r);
        } else {
            r = pos;                                      // pos < fg_this <= fgN
        }
        keep = select_rank(pf, skey, sidx, (size_t)b * M_, r);
    } else if (bgN > 0) {
        int r = (int)(u_bg[b * R_ + pos] * (float)bgN);
        r = r < 0 ? 0 : (r > bgN - 1 ? bgN - 1 : r);
        keep = select_rank(pb, skey, sidx, (size_t)b * M_ + fgN, r);
    }

    float x1, y1, x2, y2;
    if (keep < N_) {
        const float* p = all_rois + ((size_t)b * N_ + keep) * 5;
        x1 = p[1]; y1 = p[2]; x2 = p[3]; y2 = p[4];
    } else {
        const float* p = gt_boxes + ((size_t)b * G_ + (keep - N_)) * 6;
        x1 = p[0]; y1 = p[1]; x2 = p[2]; y2 = p[3];
    }

    float* rois_o = out;                       // B*R*5
    float* lab_o  = out + (size_t)B_ * R_ * 5; // B*R
    float* tid_o  = lab_o + (size_t)B_ * R_;   // B*R
    size_t o = (size_t)b * R_ + pos;
    rois_o[o * 5 + 0] = (float)b;              // column 0 overwritten with batch idx
    rois_o[o * 5 + 1] = x1;
    rois_o[o * 5 + 2] = y1;
    rois_o[o * 5 + 3] = x2;
    rois_o[o * 5 + 4] = y2;

    int ga = gt_assign[(size_t)b * M_ + keep];
    float lab = gt_boxes[((size_t)b * G_ + ga) * 6 + 4];
    float tid = gt_boxes[((size_t)b * G_ + ga) * 6 + 5];
    lab_o[o] = is_fg ? lab : 0.0f;
    tid_o[o] = is_fg ? tid : -1.0f;
}

// ---------------------------------------------------------------------------
extern "C" void kernel_launch(void* const* d_in, const int* in_sizes, int n_in,
                              void* d_out, int out_size, void* d_ws, size_t ws_size,
                              hipStream_t stream) {
    (void)in_sizes; (void)n_in; (void)out_size; (void)ws_size;
    const float* all_rois = (const float*)d_in[0];
    const float* gt_boxes = (const float*)d_in[1];
    const float* u_perm   = (const float*)d_in[2];
    const float* u_fg     = (const float*)d_in[3];
    const float* u_bg     = (const float*)d_in[4];

    char* ws = (char*)d_ws;
    size_t off = 0;
    auto alloc = [&](size_t bytes) -> void* {
        void* p = ws + off;
        off += (bytes + 255) & ~(size_t)255;
        return p;
    };
    float* max_ov    = (float*)alloc(sizeof(float) * (size_t)B_ * M_);
    int*   gt_assign = (int*)  alloc(sizeof(int)   * (size_t)B_ * M_);
    int*   hist      = (int*)  alloc(sizeof(int)   * (size_t)B_ * 2 * NBUCK);
    int*   prefix    = (int*)  alloc(sizeof(int)   * (size_t)B_ * 2 * PSTRIDE);
    int*   fill      = (int*)  alloc(sizeof(int)   * (size_t)B_ * 2 * NBUCK);
    float* skey      = (float*)alloc(sizeof(float) * (size_t)B_ * M_);
    int*   sidx      = (int*)  alloc(sizeof(int)   * (size_t)B_ * M_);

    k_zero<<<(B_ * 2 * NBUCK + 255) / 256, 256, 0, stream>>>(hist, fill, B_ * 2 * NBUCK);

    dim3 g1((M_ + 255) / 256, B_);
    k_overlap<<<g1, 256, 0, stream>>>(all_rois, gt_boxes, u_perm, max_ov, gt_assign, hist);
    k_prefix<<<B_ * 2, 32, 0, stream>>>(hist, prefix);
    k_scatter<<<g1, 256, 0, stream>>>(u_perm, max_ov, prefix, fill, skey, sidx);
    k_sample<<<B_, R_, 0, stream>>>(all_rois, gt_boxes, u_fg, u_bg,
                                    prefix, skey, sidx, gt_assign, (float*)d_out);
}